// aAsyn_GNN_70188355551848
// MI455X (gfx1250) — compile-verified
//
#include <hip/hip_runtime.h>
#include <hip/hip_bf16.h>
#include <math.h>

typedef __attribute__((ext_vector_type(2))) float v2f;
typedef __attribute__((ext_vector_type(8))) float v8f;

#define DIM 64
#define ROWS_PER_BLOCK 128
#define LDP 68    // sX row stride (floats); 272B, 16B aligned, conflict-free A reads
#define WPITCH 80 // sW2 row stride (float2); 640B == 32 banks mod 64 -> halves disjoint

// ---------------------------------------------------------------------------
// 1) softmax of comb_w rows: [4,3] -> wsoft [4,3]
// ---------------------------------------------------------------------------
__global__ void comb_softmax_kernel(const float* __restrict__ cw,
                                    float* __restrict__ wsoft) {
  int j = threadIdx.x;
  if (j < 4) {
    float a = cw[j * 3 + 0], b = cw[j * 3 + 1], c = cw[j * 3 + 2];
    float m = fmaxf(a, fmaxf(b, c));
    float ea = expf(a - m), eb = expf(b - m), ec = expf(c - m);
    float s = ea + eb + ec;
    wsoft[j * 3 + 0] = ea / s;
    wsoft[j * 3 + 1] = eb / s;
    wsoft[j * 3 + 2] = ec / s;
  }
}

// ---------------------------------------------------------------------------
// 2) fused combine + GEMM:  xw[j] = (sum_i w[j,i]*bb[1+i]) @ W_j
//    grid = (ceil(n/128), 4), block = 256 (8 wave32), fp32 WMMA 16x16x4
// ---------------------------------------------------------------------------
__global__ __launch_bounds__(256) void combine_gemm_kernel(
    const float* __restrict__ bb,     // [4][n][64]
    const float* __restrict__ wsoft,  // [4][3]
    const float* __restrict__ egoW,   // [64][64]
    const float* __restrict__ convW,  // [3][64][64]
    float* __restrict__ xw,           // [4][n][64]
    int n) {
  const int j = blockIdx.y;
  const int row0 = blockIdx.x * ROWS_PER_BLOCK;
  const int tid = threadIdx.x;

  __shared__ float sX[ROWS_PER_BLOCK][LDP];
  __shared__ v2f sW2[DIM / 2][WPITCH];  // [k-pair][col] = {W[2p][c], W[2p+1][c]}

  const float w0 = wsoft[j * 3 + 0];
  const float w1 = wsoft[j * 3 + 1];
  const float w2 = wsoft[j * 3 + 2];

  const float4* b1v = (const float4*)(bb + (size_t)1 * n * DIM);
  const float4* b2v = (const float4*)(bb + (size_t)2 * n * DIM);
  const float4* b3v = (const float4*)(bb + (size_t)3 * n * DIM);

  // stage weight matrix (row-major [k][o]) into K-pair-interleaved float2s
  const float* W = (j == 0) ? egoW : (convW + (size_t)(j - 1) * DIM * DIM);
  for (int idx = tid; idx < DIM * DIM; idx += 256) {
    int k = idx >> 6, c = idx & 63;
    ((float*)&sW2[k >> 1][c])[k & 1] = W[idx];
  }
  // stage softmax-mixed activations, zero-padded past n
  for (int idx = tid; idx < ROWS_PER_BLOCK * (DIM / 4); idx += 256) {
    int r = idx >> 4;
    int c4 = idx & 15;
    int gr = row0 + r;
    float4 v = make_float4(0.f, 0.f, 0.f, 0.f);
    if (gr < n) {
      size_t o = (size_t)gr * (DIM / 4) + c4;
      float4 x1 = b1v[o], x2 = b2v[o], x3 = b3v[o];
      v.x = w0 * x1.x + w1 * x2.x + w2 * x3.x;
      v.y = w0 * x1.y + w1 * x2.y + w2 * x3.y;
      v.z = w0 * x1.z + w1 * x2.z + w2 * x3.z;
      v.w = w0 * x1.w + w1 * x2.w + w2 * x3.w;
    }
    *(float4*)&sX[r][c4 * 4] = v;
  }
  __syncthreads();

  const int wave = tid >> 5;       // 0..7 -> 16-row tile
  const int lane = tid & 31;
  const int m0 = wave * 16;
  const int lhalf = lane >> 4;     // 0: K pair {0,1}, 1: K pair {2,3}
  const int l15 = lane & 15;

  v8f acc[4] = {};  // 4 N-tiles of 16 -> full 64 output cols

#pragma unroll 4
  for (int kk = 0; kk < DIM; kk += 4) {
    // A frag (16x4 f32): VGPR0 = K=kk (lanes 0-15) / K=kk+2 (lanes 16-31)
    // aligned 8B load: {sX[r][kk+2h], sX[r][kk+2h+1]}
    v2f a = *(const v2f*)&sX[m0 + l15][kk + 2 * lhalf];
#pragma unroll
    for (int t = 0; t < 4; ++t) {
      // B frag (4x16 f32): single ds_load_b64, pair row = kk/2 + lhalf
      v2f b = sW2[(kk >> 1) + lhalf][t * 16 + l15];
      acc[t] = __builtin_amdgcn_wmma_f32_16x16x4_f32(
          false, a, false, b, (short)0, acc[t], false, false);
    }
  }

  // C/D layout: VGPR v -> M = v + 8*lhalf, N = l15
  float* xwj = xw + (size_t)j * n * DIM;
  if (row0 + ROWS_PER_BLOCK <= n) {
    // fast path (all blocks but the last): no per-store predication
    float* base = xwj + (size_t)(row0 + m0 + 8 * lhalf) * DIM + l15;
#pragma unroll
    for (int t = 0; t < 4; ++t) {
#pragma unroll
      for (int v = 0; v < 8; ++v) {
        base[(size_t)v * DIM + t * 16] = acc[t][v];
      }
    }
  } else {
#pragma unroll
    for (int t = 0; t < 4; ++t) {
#pragma unroll
      for (int v = 0; v < 8; ++v) {
        int gr = row0 + m0 + v + 8 * lhalf;
        if (gr < n) xwj[(size_t)gr * DIM + t * 16 + l15] = acc[t][v];
      }
    }
  }
}

// ---------------------------------------------------------------------------
// 3) degree / dinv
// ---------------------------------------------------------------------------
__global__ void deg_init_kernel(float* __restrict__ deg, int n) {
  int i = blockIdx.x * 256 + threadIdx.x;
  if (i < n) deg[i] = 1.0f;  // self-loop contributes 1
}

__global__ void deg_count_kernel(const int* __restrict__ dst,
                                 float* __restrict__ deg, int nEdges) {
  int e = blockIdx.x * 256 + threadIdx.x;
  if (e < nEdges) unsafeAtomicAdd(&deg[dst[e]], 1.0f);
}

__global__ void dinv_kernel(float* __restrict__ deg, int n) {
  int i = blockIdx.x * 256 + threadIdx.x;
  if (i < n) deg[i] = rsqrtf(deg[i]);  // in place: deg -> dinv
}

// ---------------------------------------------------------------------------
// 4) agg init (self-loop term) + edge scatter
// ---------------------------------------------------------------------------
__global__ void agg_init_kernel(const float* __restrict__ dinv,
                                const float* __restrict__ xwk,
                                float* __restrict__ agg, int n) {
  size_t i = (size_t)blockIdx.x * 256 + threadIdx.x;
  if (i < (size_t)n * DIM) {
    int node = (int)(i >> 6);
    float di = dinv[node];
    agg[i] = di * di * xwk[i];
  }
}

__global__ __launch_bounds__(256) void scatter_kernel(
    const int* __restrict__ src, const int* __restrict__ dst,
    const float* __restrict__ dinv, const float* __restrict__ xwk,
    float* __restrict__ agg, int nEdges) {
  size_t t = (size_t)blockIdx.x * 256 + threadIdx.x;
  int e = (int)(t >> 4);
  int q = (int)(t & 15);  // 16 lanes * float4 = 64 features per edge
  if (e >= nEdges) return;
  int s = src[e];
  int d = dst[e];
  float c = dinv[s] * dinv[d];
  float4 xv = ((const float4*)(xwk + (size_t)s * DIM))[q];
  float* ap = agg + (size_t)d * DIM + q * 4;
  unsafeAtomicAdd(ap + 0, c * xv.x);
  unsafeAtomicAdd(ap + 1, c * xv.y);
  unsafeAtomicAdd(ap + 2, c * xv.z);
  unsafeAtomicAdd(ap + 3, c * xv.w);
}

// ---------------------------------------------------------------------------
// 5) BN statistics: sums[0..63]=sum x, sums[64..127]=sum x^2
// ---------------------------------------------------------------------------
__global__ void stats_zero_kernel(float* __restrict__ sums) {
  int i = threadIdx.x;
  if (i < 128) sums[i] = 0.0f;
}

__global__ __launch_bounds__(256) void stats_kernel(const float* __restrict__ x,
                                                    float* __restrict__ sums,
                                                    int n) {
  int f = threadIdx.x & 63;
  int g = threadIdx.x >> 6;  // 4 row-groups per block
  float s = 0.f, ss = 0.f;
  for (int r = blockIdx.x * 4 + g; r < n; r += gridDim.x * 4) {
    float v = x[(size_t)r * DIM + f];
    s += v;
    ss += v * v;
  }
  __shared__ float red[2][4][DIM];
  red[0][g][f] = s;
  red[1][g][f] = ss;
  __syncthreads();
  if (g == 0) {
    s = red[0][0][f] + red[0][1][f] + red[0][2][f] + red[0][3][f];
    ss = red[1][0][f] + red[1][1][f] + red[1][2][f] + red[1][3][f];
    unsafeAtomicAdd(&sums[f], s);
    unsafeAtomicAdd(&sums[64 + f], ss);
  }
}

__global__ void stats_final_kernel(const float* __restrict__ sums,
                                   float* __restrict__ mr, int n) {
  int f = threadIdx.x;
  if (f < 64) {
    float invN = 1.0f / (float)n;
    float m = sums[f] * invN;
    float var = sums[64 + f] * invN - m * m;
    mr[f] = m;
    mr[64 + f] = rsqrtf(var + 1e-5f);
  }
}

// ---------------------------------------------------------------------------
// 6) hidden = BN(xw0); hidden += relu(BN(agg_k))
// ---------------------------------------------------------------------------
__global__ void bn_init_kernel(const float* __restrict__ x,
                               const float* __restrict__ mr,
                               float* __restrict__ out, int n) {
  size_t i = (size_t)blockIdx.x * 256 + threadIdx.x;
  if (i < (size_t)n * DIM) {
    int f = (int)(i & 63);
    out[i] = (x[i] - mr[f]) * mr[64 + f];
  }
}

__global__ void bn_acc_relu_kernel(const float* __restrict__ x,
                                   const float* __restrict__ mr,
                                   float* __restrict__ out, int n) {
  size_t i = (size_t)blockIdx.x * 256 + threadIdx.x;
  if (i < (size_t)n * DIM) {
    int f = (int)(i & 63);
    out[i] += fmaxf(0.0f, (x[i] - mr[f]) * mr[64 + f]);
  }
}

// ---------------------------------------------------------------------------
// 7) row-wise log_softmax over 64 features, one wave32 per node
// ---------------------------------------------------------------------------
__global__ __launch_bounds__(256) void logsoftmax_kernel(float* __restrict__ out,
                                                         int n) {
  int node = blockIdx.x * 8 + (threadIdx.x >> 5);
  int lane = threadIdx.x & 31;
  if (node >= n) return;
  float* row = out + (size_t)node * DIM;
  float h0 = row[lane];
  float h1 = row[lane + 32];
  float m = fmaxf(h0, h1);
#pragma unroll
  for (int o = 16; o >= 1; o >>= 1) m = fmaxf(m, __shfl_xor(m, o, 32));
  float s = expf(h0 - m) + expf(h1 - m);
#pragma unroll
  for (int o = 16; o >= 1; o >>= 1) s += __shfl_xor(s, o, 32);
  float lse = m + logf(s);
  row[lane] = h0 - lse;
  row[lane + 32] = h1 - lse;
}

// ---------------------------------------------------------------------------
// launch
// ---------------------------------------------------------------------------
extern "C" void kernel_launch(void* const* d_in, const int* in_sizes, int n_in,
                              void* d_out, int out_size, void* d_ws,
                              size_t ws_size, hipStream_t stream) {
  const float* bb     = (const float*)d_in[0];  // [4,n,64]
  const float* comb_w = (const float*)d_in[1];  // [4,3]
  const float* egoW   = (const float*)d_in[2];  // [64,64]
  const float* convW  = (const float*)d_in[4];  // [3,64,64]
  const int*   khop   = (const int*)d_in[6];    // [3,2,E]
  float* out = (float*)d_out;

  const int n = in_sizes[0] / (4 * DIM);
  const int E = in_sizes[6] / 6;

  // workspace layout (floats)
  float* ws    = (float*)d_ws;
  float* wsoft = ws;                              // 16
  float* sums  = ws + 16;                         // 128
  float* mr    = ws + 144;                        // 128
  float* xw    = ws + 256;                        // 4*n*64
  float* agg   = xw + (size_t)4 * n * DIM;        // n*64
  float* dinv  = agg + (size_t)n * DIM;           // n

  const int nThreads = 256;
  const int gNF   = (int)(((size_t)n * DIM + nThreads - 1) / nThreads);
  const int gN    = (n + nThreads - 1) / nThreads;
  const int gE    = (E + nThreads - 1) / nThreads;
  const int gScat = (int)(((size_t)E * 16 + nThreads - 1) / nThreads);

  // 1) softmax of comb weights
  comb_softmax_kernel<<<1, 32, 0, stream>>>(comb_w, wsoft);

  // 2) fused combine + 4 GEMMs (fp32 WMMA)
  dim3 ggrid((n + ROWS_PER_BLOCK - 1) / ROWS_PER_BLOCK, 4);
  combine_gemm_kernel<<<ggrid, nThreads, 0, stream>>>(bb, wsoft, egoW, convW,
                                                      xw, n);

  // 3) ego path: hidden = BN(xw0)
  stats_zero_kernel<<<1, 128, 0, stream>>>(sums);
  stats_kernel<<<512, nThreads, 0, stream>>>(xw, sums, n);
  stats_final_kernel<<<1, 64, 0, stream>>>(sums, mr, n);
  bn_init_kernel<<<gNF, nThreads, 0, stream>>>(xw, mr, out, n);

  // 4) hops
  for (int k = 0; k < 3; ++k) {
    const int* src = khop + (size_t)k * 2 * E;
    const int* dst = src + E;
    const float* xwk = xw + (size_t)(k + 1) * n * DIM;

    deg_init_kernel<<<gN, nThreads, 0, stream>>>(dinv, n);
    deg_count_kernel<<<gE, nThreads, 0, stream>>>(dst, dinv, E);
    dinv_kernel<<<gN, nThreads, 0, stream>>>(dinv, n);

    agg_init_kernel<<<gNF, nThreads, 0, stream>>>(dinv, xwk, agg, n);
    scatter_kernel<<<gScat, nThreads, 0, stream>>>(src, dst, dinv, xwk, agg, E);

    stats_zero_kernel<<<1, 128, 0, stream>>>(sums);
    stats_kernel<<<512, nThreads, 0, stream>>>(agg, sums, n);
    stats_final_kernel<<<1, 64, 0, stream>>>(sums, mr, n);
    bn_acc_relu_kernel<<<gNF, nThreads, 0, stream>>>(agg, mr, out, n);
  }

  // 5) row-wise log_softmax
  logsoftmax_kernel<<<(n + 7) / 8, nThreads, 0, stream>>>(out, n);
}